// GNN_GAT_10857677325089
// MI455X (gfx1250) — compile-verified
//
#include <hip/hip_runtime.h>
#include <cstdint>

#define N_NODES    100000
#define N_EDGES    3200000
#define E_TOT      (N_EDGES + N_NODES)   // edges + self loops
#define IN_DIM     512
#define HEADS      4
#define C1         8
#define C2         4
#define HC1        (HEADS * C1)          // 32
#define HC2        (HEADS * C2)          // 16
#define OUT_DIM    10
#define NUM_GRAPHS 64
#define NEG_SLOPE  0.2f

typedef float v2f __attribute__((ext_vector_type(2)));
typedef float v8f __attribute__((ext_vector_type(8)));

// ---- monotone float<->int mapping for native integer atomicMax ----
__device__ __forceinline__ int f32_ordered(float f) {
    int i = __float_as_int(f);
    return i >= 0 ? i : (i ^ 0x7FFFFFFF);
}
__device__ __forceinline__ float ordered_f32(int i) {
    return __int_as_float(i >= 0 ? i : (i ^ 0x7FFFFFFF));
}

// ============================================================
// GEMM1: h1[N,32] = x[N,512] @ W1[512,32]  (fp32 WMMA 16x16x4)
// one wave -> 16-row M tile, both 16-col N tiles (A reused)
// ============================================================
__global__ void __launch_bounds__(256) gat_gemm1(const float* __restrict__ x,
                                                 const float* __restrict__ W1,
                                                 float* __restrict__ h1) {
    int wave = (int)(blockIdx.x * (blockDim.x >> 5) + (threadIdx.x >> 5));
    int lane = threadIdx.x & 31;
    if (wave >= N_NODES / 16) return;            // uniform per wave
    int half = lane >> 4;                        // 0: K=0,1  1: K=2,3
    int l    = lane & 15;
    int m0   = wave * 16;
    const float* xrow = x + (size_t)(m0 + l) * IN_DIM;
    int koff = half * 2;

    v8f c0 = {}; v8f c1 = {};
    for (int k = 0; k < IN_DIM; k += 4) {
        v2f a;
        a.x = xrow[k + koff];
        a.y = xrow[k + koff + 1];
        const float* wrow = W1 + (size_t)(k + koff) * HC1;
        v2f b0, b1;
        b0.x = wrow[l];            b0.y = wrow[HC1 + l];          // cols 0..15
        b1.x = wrow[16 + l];       b1.y = wrow[HC1 + 16 + l];     // cols 16..31
        c0 = __builtin_amdgcn_wmma_f32_16x16x4_f32(false, a, false, b0, (short)0, c0, false, false);
        c1 = __builtin_amdgcn_wmma_f32_16x16x4_f32(false, a, false, b1, (short)0, c1, false, false);
    }
    int rbase = m0 + half * 8;                   // D: vgpr r -> M = r (+8 for hi half)
#pragma unroll
    for (int r = 0; r < 8; ++r) {
        h1[(size_t)(rbase + r) * HC1 + l]      = c0[r];
        h1[(size_t)(rbase + r) * HC1 + 16 + l] = c1[r];
    }
}

// ============================================================
// GEMM2: h2[N,16] = h1[N,32] @ W2[32,16]
// ============================================================
__global__ void __launch_bounds__(256) gat_gemm2(const float* __restrict__ h1,
                                                 const float* __restrict__ W2,
                                                 float* __restrict__ h2) {
    int wave = (int)(blockIdx.x * (blockDim.x >> 5) + (threadIdx.x >> 5));
    int lane = threadIdx.x & 31;
    if (wave >= N_NODES / 16) return;
    int half = lane >> 4;
    int l    = lane & 15;
    int m0   = wave * 16;
    const float* arow = h1 + (size_t)(m0 + l) * HC1;
    int koff = half * 2;

    v8f c = {};
#pragma unroll
    for (int k = 0; k < HC1; k += 4) {
        v2f a;
        a.x = arow[k + koff];
        a.y = arow[k + koff + 1];
        const float* wrow = W2 + (size_t)(k + koff) * HC2;
        v2f b;
        b.x = wrow[l];
        b.y = wrow[HC2 + l];
        c = __builtin_amdgcn_wmma_f32_16x16x4_f32(false, a, false, b, (short)0, c, false, false);
    }
    int rbase = m0 + half * 8;
#pragma unroll
    for (int r = 0; r < 8; ++r)
        h2[(size_t)(rbase + r) * HC2 + l] = c[r];
}

// ============================================================
// per-node attention scores: es[n,h] = sum_c h[n,h,c]*a_src[h,c]
// ============================================================
template <int C>
__global__ void node_scores(const float* __restrict__ h,
                            const float* __restrict__ a_src,
                            const float* __restrict__ a_dst,
                            float* __restrict__ es, float* __restrict__ ed) {
    int n = blockIdx.x * blockDim.x + threadIdx.x;
    if (n >= N_NODES) return;
    const float* hn = h + (size_t)n * HEADS * C;
#pragma unroll
    for (int hd = 0; hd < HEADS; ++hd) {
        float ss = 0.f, sd = 0.f;
#pragma unroll
        for (int c = 0; c < C; ++c) {
            float v = hn[hd * C + c];
            ss += v * a_src[hd * C + c];
            sd += v * a_dst[hd * C + c];
        }
        es[n * HEADS + hd] = ss;
        ed[n * HEADS + hd] = sd;
    }
}

// ---- per-conv scratch init ----
__global__ void init_conv(int* __restrict__ mI, float* __restrict__ den,
                          float* __restrict__ agg, int aggdim) {
    long i = (long)blockIdx.x * blockDim.x + threadIdx.x;
    if (i < (long)N_NODES * HEADS) { mI[i] = (int)0x80000000; den[i] = 0.f; }
    if (i < (long)N_NODES * aggdim) agg[i] = 0.f;
}

__device__ __forceinline__ void edge_pair(int e, const int* __restrict__ ei, int& s, int& d) {
    if (e < N_EDGES) { s = ei[e]; d = ei[N_EDGES + e]; }
    else             { s = d = e - N_EDGES; }
}

// pass 1: segment max of leaky(e_src[s]+e_dst[d])
__global__ void edge_max(const int* __restrict__ ei, const float* __restrict__ es,
                         const float* __restrict__ ed, int* __restrict__ mI) {
    int e = blockIdx.x * blockDim.x + threadIdx.x;
    if (e >= E_TOT) return;
    int s, d; edge_pair(e, ei, s, d);
#pragma unroll
    for (int hd = 0; hd < HEADS; ++hd) {
        float v = es[s * HEADS + hd] + ed[d * HEADS + hd];
        v = v > 0.f ? v : NEG_SLOPE * v;
        atomicMax(&mI[d * HEADS + hd], f32_ordered(v));
    }
}

// pass 2: denom = segment sum of exp(e - m[dst])
__global__ void edge_sum(const int* __restrict__ ei, const float* __restrict__ es,
                         const float* __restrict__ ed, const int* __restrict__ mI,
                         float* __restrict__ den) {
    int e = blockIdx.x * blockDim.x + threadIdx.x;
    if (e >= E_TOT) return;
    int s, d; edge_pair(e, ei, s, d);
#pragma unroll
    for (int hd = 0; hd < HEADS; ++hd) {
        float v = es[s * HEADS + hd] + ed[d * HEADS + hd];
        v = v > 0.f ? v : NEG_SLOPE * v;
        float m = ordered_f32(mI[d * HEADS + hd]);
        atomicAdd(&den[d * HEADS + hd], __expf(v - m));
    }
}

// pass 3: agg[dst] += h[src] * alpha
template <int C>
__global__ void edge_agg(const int* __restrict__ ei, const float* __restrict__ es,
                         const float* __restrict__ ed, const int* __restrict__ mI,
                         const float* __restrict__ den, const float* __restrict__ h,
                         float* __restrict__ agg) {
    int e = blockIdx.x * blockDim.x + threadIdx.x;
    if (e >= E_TOT) return;
    int s, d; edge_pair(e, ei, s, d);
#pragma unroll
    for (int hd = 0; hd < HEADS; ++hd) {
        float v = es[s * HEADS + hd] + ed[d * HEADS + hd];
        v = v > 0.f ? v : NEG_SLOPE * v;
        float m = ordered_f32(mI[d * HEADS + hd]);
        float alpha = __expf(v - m) / den[d * HEADS + hd];
        const float* hs = h + (size_t)s * HEADS * C + hd * C;
        float* ad = agg + (size_t)d * HEADS * C + hd * C;
#pragma unroll
        for (int c = 0; c < C; ++c)
            atomicAdd(&ad[c], hs[c] * alpha);
    }
}

// relu(agg + b) -> out (written over the h buffer)
__global__ void bias_relu(const float* __restrict__ agg, const float* __restrict__ b,
                          float* __restrict__ out, int dim) {
    long i = (long)blockIdx.x * blockDim.x + threadIdx.x;
    if (i >= (long)N_NODES * dim) return;
    float v = agg[i] + b[i % dim];
    out[i] = v > 0.f ? v : 0.f;
}

// ---- pooling + final linear ----
__global__ void init_pool(float* __restrict__ psum, float* __restrict__ pcnt) {
    int i = blockIdx.x * blockDim.x + threadIdx.x;
    if (i < NUM_GRAPHS * HC2) psum[i] = 0.f;
    if (i < NUM_GRAPHS) pcnt[i] = 0.f;
}

__global__ void pool_accum(const float* __restrict__ h, const int* __restrict__ batch,
                           float* __restrict__ psum, float* __restrict__ pcnt) {
    int n = blockIdx.x * blockDim.x + threadIdx.x;
    if (n >= N_NODES) return;
    int g = batch[n];
#pragma unroll
    for (int j = 0; j < HC2; ++j)
        atomicAdd(&psum[g * HC2 + j], h[(size_t)n * HC2 + j]);
    atomicAdd(&pcnt[g], 1.0f);
}

__global__ void final_fc(const float* __restrict__ psum, const float* __restrict__ pcnt,
                         const float* __restrict__ Wfc, const float* __restrict__ bfc,
                         float* __restrict__ out) {
    int i = blockIdx.x * blockDim.x + threadIdx.x;
    if (i >= NUM_GRAPHS * OUT_DIM) return;
    int g = i / OUT_DIM, o = i % OUT_DIM;
    float cnt = pcnt[g]; cnt = cnt > 1.f ? cnt : 1.f;
    float acc = bfc[o];
#pragma unroll
    for (int j = 0; j < HC2; ++j)
        acc += (psum[g * HC2 + j] / cnt) * Wfc[j * OUT_DIM + o];
    out[i] = acc;
}

extern "C" void kernel_launch(void* const* d_in, const int* in_sizes, int n_in,
                              void* d_out, int out_size, void* d_ws, size_t ws_size,
                              hipStream_t stream) {
    const float* x      = (const float*)d_in[0];
    const int*   ei     = (const int*)d_in[1];
    const int*   batch  = (const int*)d_in[2];
    const float* W1     = (const float*)d_in[3];
    const float* a1s    = (const float*)d_in[4];
    const float* a1d    = (const float*)d_in[5];
    const float* b1     = (const float*)d_in[6];
    const float* W2     = (const float*)d_in[7];
    const float* a2s    = (const float*)d_in[8];
    const float* a2d    = (const float*)d_in[9];
    const float* b2     = (const float*)d_in[10];
    const float* Wfc    = (const float*)d_in[11];
    const float* bfc    = (const float*)d_in[12];
    float* out = (float*)d_out;

    // workspace layout (floats)
    float* ws   = (float*)d_ws;
    float* h1   = ws;                         // N*32
    float* agg1 = h1   + (size_t)N_NODES * HC1;
    float* es1  = agg1 + (size_t)N_NODES * HC1;   // N*4
    float* ed1  = es1  + (size_t)N_NODES * HEADS;
    float* den1 = ed1  + (size_t)N_NODES * HEADS;
    int*   m1i  = (int*)(den1 + (size_t)N_NODES * HEADS);
    float* h2   = (float*)(m1i + (size_t)N_NODES * HEADS); // N*16
    float* agg2 = h2   + (size_t)N_NODES * HC2;
    float* es2  = agg2 + (size_t)N_NODES * HC2;
    float* ed2  = es2  + (size_t)N_NODES * HEADS;
    float* den2 = ed2  + (size_t)N_NODES * HEADS;
    int*   m2i  = (int*)(den2 + (size_t)N_NODES * HEADS);
    float* psum = (float*)(m2i + (size_t)N_NODES * HEADS);
    float* pcnt = psum + NUM_GRAPHS * HC2;

    const int TB = 256;
    const int gNodes = (N_NODES + TB - 1) / TB;
    const int gEdges = (E_TOT + TB - 1) / TB;
    const int gWaves = ((N_NODES / 16) + 7) / 8;      // 8 waves (256 thr) / block
    const int gN32   = (N_NODES * HC1 + TB - 1) / TB;
    const int gN16   = (N_NODES * HC2 + TB - 1) / TB;

    // ---------------- conv1 ----------------
    gat_gemm1<<<gWaves, TB, 0, stream>>>(x, W1, h1);
    node_scores<C1><<<gNodes, TB, 0, stream>>>(h1, a1s, a1d, es1, ed1);
    init_conv<<<gN32, TB, 0, stream>>>(m1i, den1, agg1, HC1);
    edge_max<<<gEdges, TB, 0, stream>>>(ei, es1, ed1, m1i);
    edge_sum<<<gEdges, TB, 0, stream>>>(ei, es1, ed1, m1i, den1);
    edge_agg<C1><<<gEdges, TB, 0, stream>>>(ei, es1, ed1, m1i, den1, h1, agg1);
    bias_relu<<<gN32, TB, 0, stream>>>(agg1, b1, h1, HC1);   // h1 := relu(conv1 out)

    // ---------------- conv2 ----------------
    gat_gemm2<<<gWaves, TB, 0, stream>>>(h1, W2, h2);
    node_scores<C2><<<gNodes, TB, 0, stream>>>(h2, a2s, a2d, es2, ed2);
    init_conv<<<gN16, TB, 0, stream>>>(m2i, den2, agg2, HC2);
    edge_max<<<gEdges, TB, 0, stream>>>(ei, es2, ed2, m2i);
    edge_sum<<<gEdges, TB, 0, stream>>>(ei, es2, ed2, m2i, den2);
    edge_agg<C2><<<gEdges, TB, 0, stream>>>(ei, es2, ed2, m2i, den2, h2, agg2);
    bias_relu<<<gN16, TB, 0, stream>>>(agg2, b2, h2, HC2);   // h2 := relu(conv2 out)

    // ---------------- pool + fc ----------------
    init_pool<<<(NUM_GRAPHS * HC2 + TB - 1) / TB, TB, 0, stream>>>(psum, pcnt);
    pool_accum<<<gNodes, TB, 0, stream>>>(h2, batch, psum, pcnt);
    final_fc<<<(NUM_GRAPHS * OUT_DIM + TB - 1) / TB, TB, 0, stream>>>(psum, pcnt, Wfc, bfc, out);
}